// PositionwiseFeedForward_51994874085871
// MI455X (gfx1250) — compile-verified
//
#include <hip/hip_runtime.h>
#include <cmath>

typedef __attribute__((ext_vector_type(2))) float v2f;
typedef __attribute__((ext_vector_type(8))) float v8f;

#define NUM_DENSE 2
#define NBASIS    20
#define HDIM      64
#define CDIM      32
#define BATCH     2
#define NPTS      192
#define TILE_A    16
#define NSPLIT    12
#define BCHUNK    (NPTS / NSPLIT)   // 16 b's per workgroup
#define HSTR      68                // LDS stride for h-tiles: bank = 4m+2hf -> conflict-free A reads
#define WSTR      80                // LDS stride for weights: 2*WSTR % 64 == 32 -> conflict-free B reads

__device__ __forceinline__ float cos_basis(float d, int k) {
    // radii = linspace(0,10,20); step = 10/19
    const float step = 10.0f / 19.0f;
    float y = (d - (float)k * step) * (19.0f / 10.0f);
    y = fminf(fmaxf(y, -1.0f), 1.0f);
    return __cosf(1.57079632679489662f * y);   // cos(pi/2 * y) via v_cos_f32
}

// ---------------------------------------------------------------------------
// G[z,b,h,i] = sum_j W2[h, i*C+j] * F[z,b,j]
// ---------------------------------------------------------------------------
__global__ void g_kernel(const float* __restrict__ F, const float* __restrict__ W2,
                         float* __restrict__ G) {
    int gid = blockIdx.x * blockDim.x + threadIdx.x;
    if (gid >= BATCH * NPTS * HDIM * CDIM) return;
    const int i  = gid % CDIM;
    const int h  = (gid / CDIM) % HDIM;
    const int zb = gid / (CDIM * HDIM);
    const float* f = F + (size_t)zb * CDIM;
    const float* w = W2 + (size_t)h * (CDIM * CDIM) + (size_t)i * CDIM;
    float s = 0.0f;
    #pragma unroll
    for (int j = 0; j < CDIM; ++j) s += w[j] * f[j];
    G[gid] = s;
}

// Fsum[z,j] = sum_b F[z,b,j]   (for the b2 bias term)
__global__ void fsum_kernel(const float* __restrict__ F, float* __restrict__ Fsum) {
    int t = blockIdx.x * blockDim.x + threadIdx.x;
    if (t >= BATCH * CDIM) return;
    const int z = t / CDIM, j = t % CDIM;
    float s = 0.0f;
    for (int b = 0; b < NPTS; ++b) s += F[((size_t)z * NPTS + b) * CDIM + j];
    Fsum[t] = s;
}

__global__ void zero_kernel(float* __restrict__ p, int n) {
    int t = blockIdx.x * blockDim.x + threadIdx.x;
    if (t < n) p[t] = 0.0f;
}

// out = (acc + b2-bias-term) * n^-0.5 * mask
__global__ void finalize_kernel(float* __restrict__ out, const float* __restrict__ b2,
                                const float* __restrict__ Fsum, const float* __restrict__ mask) {
    int t = blockIdx.x * blockDim.x + threadIdx.x;
    if (t >= BATCH * NPTS * CDIM) return;
    const int i  = t % CDIM;
    const int za = t / CDIM;
    const int z  = za / NPTS;
    float bias = 0.0f;
    #pragma unroll
    for (int j = 0; j < CDIM; ++j) bias += b2[i * CDIM + j] * Fsum[z * CDIM + j];
    const float scale = 0.07216878364870322f;  // 1/sqrt(192)
    out[t] = (out[t] + bias) * scale * mask[za];
}

// ---------------------------------------------------------------------------
// Fused per-edge MLP + contraction, fp32 WMMA 16x16x4.
// One wave owns a 16-wide a-tile and BCHUNK b's; accumulates OUT[16a x 32i].
// ---------------------------------------------------------------------------
__global__ __launch_bounds__(32)
void conv_edge_wmma(const float* __restrict__ xyz,
                    const float* __restrict__ W0, const float* __restrict__ b0,
                    const float* __restrict__ W1, const float* __restrict__ b1,
                    const float* __restrict__ G,
                    float* __restrict__ outacc)
{
    __shared__ float sW0[NBASIS * WSTR];
    __shared__ float sW1[HDIM * WSTR];
    __shared__ float sb0[HDIM];
    __shared__ float sb1[HDIM];
    __shared__ float sH1[TILE_A * HSTR];
    __shared__ float sH2[TILE_A * HSTR];

    const int lane = threadIdx.x;   // wave32
    const int m    = lane & 15;
    const int hf   = lane >> 4;     // lane half: selects K pair in fp32 WMMA fragments

    const int bid   = blockIdx.x;
    const int z     = bid / ((NPTS / TILE_A) * NSPLIT);
    const int rem   = bid % ((NPTS / TILE_A) * NSPLIT);
    const int atile = rem / NSPLIT;
    const int split = rem % NSPLIT;

    for (int idx = lane; idx < NBASIS * HDIM; idx += 32)
        sW0[(idx / HDIM) * WSTR + (idx % HDIM)] = W0[idx];
    for (int idx = lane; idx < HDIM * HDIM; idx += 32)
        sW1[(idx / HDIM) * WSTR + (idx % HDIM)] = W1[idx];
    for (int idx = lane; idx < HDIM; idx += 32) {
        sb0[idx] = b0[idx];
        sb1[idx] = b1[idx];
    }
    __syncthreads();

    const int a = atile * TILE_A + m;
    const float ax = xyz[((size_t)z * NPTS + a) * 3 + 0];
    const float ay = xyz[((size_t)z * NPTS + a) * 3 + 1];
    const float az = xyz[((size_t)z * NPTS + a) * 3 + 2];

    v8f acc0 = {0.f,0.f,0.f,0.f,0.f,0.f,0.f,0.f};
    v8f acc1 = {0.f,0.f,0.f,0.f,0.f,0.f,0.f,0.f};

    for (int bi = 0; bi < BCHUNK; ++bi) {
        const int bb = split * BCHUNK + bi;
        const float bx = xyz[((size_t)z * NPTS + bb) * 3 + 0];
        const float by = xyz[((size_t)z * NPTS + bb) * 3 + 1];
        const float bz = xyz[((size_t)z * NPTS + bb) * 3 + 2];
        const float dx = ax - bx, dy = ay - by, dz = az - bz;
        const float d = sqrtf(dx*dx + dy*dy + dz*dz + 1e-12f);

        // A-fragments of the radial basis [16 edges x 20], 5 K-steps.
        // fp32 A layout: lane<16 holds (M=m, K=4s,4s+1); lane>=16 holds (M=m, K=4s+2,4s+3)
        v2f ra[5];
        #pragma unroll
        for (int s = 0; s < 5; ++s) {
            const int k0 = 4 * s + 2 * hf;
            ra[s].x = cos_basis(d, k0);
            ra[s].y = cos_basis(d, k0 + 1);
        }

        // GEMM1: h1 = relu(rb @ W0 + b0)  -> stage to LDS in row-major (A layout for next GEMM)
        #pragma unroll
        for (int ht = 0; ht < 4; ++ht) {
            const int col = ht * 16 + m;
            const float bias = sb0[col];
            v8f c = {bias,bias,bias,bias,bias,bias,bias,bias};
            #pragma unroll
            for (int s = 0; s < 5; ++s) {
                const int k0 = 4 * s + 2 * hf;
                v2f bm;
                bm.x = sW0[k0 * WSTR + col];
                bm.y = sW0[(k0 + 1) * WSTR + col];
                c = __builtin_amdgcn_wmma_f32_16x16x4_f32(false, ra[s], false, bm,
                                                          (short)0, c, false, false);
            }
            #pragma unroll
            for (int rr = 0; rr < 8; ++rr)
                sH1[(rr + 8 * hf) * HSTR + col] = fmaxf(c[rr], 0.0f);
        }
        __syncthreads();

        // GEMM2: h2 = relu(h1 @ W1 + b1)
        #pragma unroll
        for (int ht = 0; ht < 4; ++ht) {
            const int col = ht * 16 + m;
            const float bias = sb1[col];
            v8f c = {bias,bias,bias,bias,bias,bias,bias,bias};
            #pragma unroll
            for (int s = 0; s < 16; ++s) {
                const int k0 = 4 * s + 2 * hf;
                v2f am, bm;
                am.x = sH1[m * HSTR + k0];
                am.y = sH1[m * HSTR + k0 + 1];
                bm.x = sW1[k0 * WSTR + col];
                bm.y = sW1[(k0 + 1) * WSTR + col];
                c = __builtin_amdgcn_wmma_f32_16x16x4_f32(false, am, false, bm,
                                                          (short)0, c, false, false);
            }
            #pragma unroll
            for (int rr = 0; rr < 8; ++rr)
                sH2[(rr + 8 * hf) * HSTR + col] = fmaxf(c[rr], 0.0f);
        }
        __syncthreads();

        // GEMM3: OUT[16a x 32i] += h2[16x64] @ G[z,bb][64x32]
        const float* Gb = G + ((size_t)(z * NPTS + bb)) * HDIM * CDIM;
        #pragma unroll
        for (int s = 0; s < 16; ++s) {
            const int k0 = 4 * s + 2 * hf;
            v2f am, b0m, b1m;
            am.x  = sH2[m * HSTR + k0];
            am.y  = sH2[m * HSTR + k0 + 1];
            b0m.x = Gb[k0 * CDIM + m];
            b0m.y = Gb[(k0 + 1) * CDIM + m];
            b1m.x = Gb[k0 * CDIM + 16 + m];
            b1m.y = Gb[(k0 + 1) * CDIM + 16 + m];
            acc0 = __builtin_amdgcn_wmma_f32_16x16x4_f32(false, am, false, b0m,
                                                         (short)0, acc0, false, false);
            acc1 = __builtin_amdgcn_wmma_f32_16x16x4_f32(false, am, false, b1m,
                                                         (short)0, acc1, false, false);
        }
        __syncthreads();
    }

    // Accumulate partial OUT tile (b-split across workgroups) with native f32 atomics.
    #pragma unroll
    for (int rr = 0; rr < 8; ++rr) {
        const int row = rr + 8 * hf;
        float* p = &outacc[((size_t)(z * NPTS + atile * TILE_A + row)) * CDIM + m];
        __hip_atomic_fetch_add(p,      acc0[rr], __ATOMIC_RELAXED, __HIP_MEMORY_SCOPE_AGENT);
        __hip_atomic_fetch_add(p + 16, acc1[rr], __ATOMIC_RELAXED, __HIP_MEMORY_SCOPE_AGENT);
    }
}

// ---------------------------------------------------------------------------
extern "C" void kernel_launch(void* const* d_in, const int* in_sizes, int n_in,
                              void* d_out, int out_size, void* d_ws, size_t ws_size,
                              hipStream_t stream) {
    (void)in_sizes; (void)n_in; (void)out_size; (void)ws_size;
    const float* features = (const float*)d_in[0];
    const float* xyz      = (const float*)d_in[1];
    const float* mask     = (const float*)d_in[2];
    const float* W0       = (const float*)d_in[3];
    const float* b0       = (const float*)d_in[4];
    const float* W1       = (const float*)d_in[5];
    const float* b1       = (const float*)d_in[6];
    const float* W2       = (const float*)d_in[7];
    const float* b2       = (const float*)d_in[8];
    float* out = (float*)d_out;

    float* G    = (float*)d_ws;                               // B*N*H*C  (~3 MB)
    float* F1   = G    + (size_t)BATCH * NPTS * HDIM * CDIM;  // B*N*C
    float* Fsum = F1   + (size_t)BATCH * NPTS * CDIM;         // B*C

    const int nG   = BATCH * NPTS * HDIM * CDIM;
    const int nOut = BATCH * NPTS * CDIM;

    for (int l = 0; l < NUM_DENSE; ++l) {
        const float* Fin  = (l == 0) ? features : F1;
        float*       Fout = (l == NUM_DENSE - 1) ? out : F1;
        const float* W0l = W0 + (size_t)l * NBASIS * HDIM;
        const float* b0l = b0 + (size_t)l * HDIM;
        const float* W1l = W1 + (size_t)l * HDIM * HDIM;
        const float* b1l = b1 + (size_t)l * HDIM;
        const float* W2l = W2 + (size_t)l * HDIM * CDIM * CDIM;
        const float* b2l = b2 + (size_t)l * CDIM * CDIM;

        fsum_kernel<<<1, 64, 0, stream>>>(Fin, Fsum);
        g_kernel<<<(nG + 255) / 256, 256, 0, stream>>>(Fin, W2l, G);
        zero_kernel<<<(nOut + 255) / 256, 256, 0, stream>>>(Fout, nOut);
        conv_edge_wmma<<<BATCH * (NPTS / TILE_A) * NSPLIT, 32, 0, stream>>>(
            xyz, W0l, b0l, W1l, b1l, G, Fout);
        finalize_kernel<<<(nOut + 255) / 256, 256, 0, stream>>>(Fout, b2l, Fsum, mask);
    }
}